// NghSampler2_9921374453828
// MI455X (gfx1250) — compile-verified
//
#include <hip/hip_runtime.h>
#include <hip/hip_bf16.h>
#include <stdint.h>

// ---------------------------------------------------------------------------
// NghSampler2 for MI455X (gfx1250, wave32).
//  prep_kernel : gather f1 / distractors, pos/neg dot products (VALU+shfl)
//  dscore_wmma : 3136x3136x128 f32 GEMM via v_wmma_f32_16x16x4_f32
//                with async-to-LDS B-panel staging (ASYNCcnt path)
//  gt_fill     : ground-truth matrix fill (NT stores)
// ---------------------------------------------------------------------------

typedef __attribute__((ext_vector_type(2))) float v2f;
typedef __attribute__((ext_vector_type(8))) float v8f;

#define NQ        3136          // B * 28 * 28 query points (== distractor count)
#define NCOLS     3217          // 1 (pos) + 80 (neg) + 3136 (distractors)
#define CC        128
#define HH        256
#define WW        256
#define HWSZ      (256 * 256)
#define G1        28
#define GRID_PER_B 784
#define NPOS      29
#define NNEG      80
#define QT        196           // NQ / 16  (M tiles)
#define DBLK      49            // NQ / 64  (N blocks of 64)
#define QBLK      25            // ceil(196 / 8) q-tiles per block row
#define BROW      132           // padded LDS row length (floats) -> bank-conflict-free

// pos offsets: i^2 + j^2 <= 9 (j outer, i inner, ascending) -> 29 entries
__constant__ int c_posx[NPOS] = {
    0, -2,-1,0,1,2, -2,-1,0,1,2, -3,-2,-1,0,1,2,3, -2,-1,0,1,2, -2,-1,0,1,2, 0};
__constant__ int c_posy[NPOS] = {
    -3, -2,-2,-2,-2,-2, -1,-1,-1,-1,-1, 0,0,0,0,0,0,0, 1,1,1,1,1, 2,2,2,2,2, 3};

// neg offsets: 25 <= i^2 + j^2 <= 49 -> 80 entries
__constant__ int c_negx[NNEG] = {
    0,
    -3,-2,-1,0,1,2,3,
    -4,-3,-2,-1,0,1,2,3,4,
    -5,-4,-3,3,4,5,
    -6,-5,-4,4,5,6,
    -6,-5,5,6,
    -6,-5,5,6,
    -7,-6,-5,5,6,7,
    -6,-5,5,6,
    -6,-5,5,6,
    -6,-5,-4,4,5,6,
    -5,-4,-3,3,4,5,
    -4,-3,-2,-1,0,1,2,3,4,
    -3,-2,-1,0,1,2,3,
    0};
__constant__ int c_negy[NNEG] = {
    -7,
    -6,-6,-6,-6,-6,-6,-6,
    -5,-5,-5,-5,-5,-5,-5,-5,-5,
    -4,-4,-4,-4,-4,-4,
    -3,-3,-3,-3,-3,-3,
    -2,-2,-2,-2,
    -1,-1,-1,-1,
    0,0,0,0,0,0,
    1,1,1,1,
    2,2,2,2,
    3,3,3,3,3,3,
    4,4,4,4,4,4,
    5,5,5,5,5,5,5,5,5,
    6,6,6,6,6,6,6,
    7};

// ---------------------------------------------------------------------------
// Kernel 1: one wave32 per query point.
// ---------------------------------------------------------------------------
__global__ void __launch_bounds__(256)
prep_kernel(const float* __restrict__ feat1, const float* __restrict__ feat2,
            const float* __restrict__ conf1, const float* __restrict__ conf2,
            const float* __restrict__ aflow,
            float* __restrict__ f1g, float* __restrict__ distg,
            int2* __restrict__ xy2ws,
            float* __restrict__ scores, float* __restrict__ mask_out,
            float* __restrict__ qconf_out)
{
    const int lane = threadIdx.x & 31;
    const int q = blockIdx.x * 8 + (threadIdx.x >> 5);
    if (q >= NQ) return;

    const int b  = q / GRID_PER_B;
    const int r  = q - b * GRID_PER_B;
    const int iy = r / G1;
    const int ix = r - iy * G1;
    const int y1 = 16 + 8 * iy;
    const int x1 = 16 + 8 * ix;
    const int pix = y1 * WW + x1;

    // lane handles channels lane, lane+32, lane+64, lane+96
    const float* f1b = feat1 + (b * CC + lane) * HWSZ;
    const float* f2b = feat2 + (b * CC + lane) * HWSZ;

    float f1v[4];
#pragma unroll
    for (int j = 0; j < 4; ++j) {
        f1v[j] = f1b[j * 32 * HWSZ + pix];
        f1g[q * CC + lane + 32 * j]   = f1v[j];                   // packed A panel
        distg[q * CC + lane + 32 * j] = f2b[j * 32 * HWSZ + pix]; // packed B panel
    }

    const float ax = aflow[(b * 2 + 0) * HWSZ + pix];
    const float ay = aflow[(b * 2 + 1) * HWSZ + pix];
    const int X2 = (int)(ax + 0.5f);   // trunc toward zero, matches astype(int32)
    const int Y2 = (int)(ay + 0.5f);
    const bool inb = (X2 >= 0) && (Y2 >= 0) && (X2 < WW) && (Y2 < HH);

    // ----- positive ring: max + argmax over 29 dots -----
    float best = -3.0e38f;
    int bestI = 0;
    for (int p = 0; p < NPOS; ++p) {
        int xx = min(max(X2 + c_posx[p], 0), WW - 1);
        int yy = min(max(Y2 + c_posy[p], 0), HH - 1);
        int o = yy * WW + xx;
        float s = 0.0f;
#pragma unroll
        for (int j = 0; j < 4; ++j) s += f1v[j] * f2b[j * 32 * HWSZ + o];
#pragma unroll
        for (int sh = 16; sh > 0; sh >>= 1) s += __shfl_xor(s, sh, 32);
        if (s > best) { best = s; bestI = p; }   // strict > keeps first argmax
    }

    // ----- negative ring: 80 dots -> scores columns 1..80 -----
    for (int nn = 0; nn < NNEG; ++nn) {
        int xx = min(max(X2 + c_negx[nn], 0), WW - 1);
        int yy = min(max(Y2 + c_negy[nn], 0), HH - 1);
        int o = yy * WW + xx;
        float s = 0.0f;
#pragma unroll
        for (int j = 0; j < 4; ++j) s += f1v[j] * f2b[j * 32 * HWSZ + o];
#pragma unroll
        for (int sh = 16; sh > 0; sh >>= 1) s += __shfl_xor(s, sh, 32);
        if (lane == 0) scores[q * NCOLS + 1 + nn] = s;
    }

    if (lane == 0) {
        scores[q * NCOLS] = best;  // pscores (max over pos ring)
        int sx = min(max(X2 + c_posx[bestI], 0), WW - 1);
        int sy = min(max(Y2 + c_posy[bestI], 0), HH - 1);
        qconf_out[q] = 0.5f * (conf1[b * HWSZ + pix] + conf2[b * HWSZ + sy * WW + sx]);
        mask_out[q] = inb ? 1.0f : 0.0f;
        xy2ws[q] = make_int2(X2, Y2);
    }
}

// ---------------------------------------------------------------------------
// Kernel 2: dscores = f1g @ distg^T (3136 x 3136 x 128) via f32 WMMA.
//
// Block = 8 waves, computes a 128(M) x 64(N) super-tile:
//   - all 256 threads stage the 64x128 B panel into LDS with
//     GLOBAL_LOAD_ASYNC_TO_LDS_B128 (ASYNCcnt), s_wait_asynccnt 0, barrier
//   - each wave owns one 16-row q-tile and 4 accumulators (16x64 strip):
//     per k-step: 1 global b64 A-fragment (reused 4x) + 4 ds b64 B-fragments
// LDS rows padded to 132 floats: fragment banks = (4*row + col) mod 64,
// all 32 lanes hit distinct banks -> conflict-free ds_load_b64.
// ---------------------------------------------------------------------------
__global__ void __launch_bounds__(256)
dscore_wmma(const float* __restrict__ f1g, const float* __restrict__ distg,
            const int2* __restrict__ xy2ws, float* __restrict__ scores)
{
    __shared__ __align__(16) float Bs[64][BROW];

    const int lane = threadIdx.x & 31;
    const int wave = threadIdx.x >> 5;
    const int qb = blockIdx.x / DBLK;          // 0..24
    const int db = blockIdx.x - qb * DBLK;     // 0..48
    const int d0 = db * 64;

    // ---- cooperative async B-panel stage: 2048 x b128 = 32 KB ----
    {
        const int tid = threadIdx.x;
        const uint32_t lds_base = (uint32_t)(uintptr_t)&Bs[0][0];
#pragma unroll
        for (int i = 0; i < 8; ++i) {
            const int ch  = tid + i * 256;       // 0..2047
            const int row = ch >> 5;             // 0..63
            const int c4  = (ch & 31) * 4;       // 0,4,...,124
            const uint32_t laddr = lds_base + (uint32_t)((row * BROW + c4) * 4);
            const uint64_t gaddr =
                (uint64_t)(uintptr_t)(distg + (size_t)(d0 + row) * CC + c4);
            asm volatile("global_load_async_to_lds_b128 %0, %1, off"
                         :: "v"(laddr), "v"(gaddr) : "memory");
        }
        asm volatile("s_wait_asynccnt 0x0" ::: "memory");
    }
    __syncthreads();   // every wave reaches this (incl. tail-idle waves)

    const int qt = qb * 8 + wave;              // wave-uniform
    if (qt < QT) {                             // whole-wave predicate: EXEC all-1s inside
        const int half = lane >> 4;
        const int n15  = lane & 15;

        const float* Ap = f1g + (qt * 16 + n15) * CC + 2 * half;

        v8f acc[4] = {};
#pragma unroll 8
        for (int kb = 0; kb < 32; ++kb) {
            v2f a = *(const v2f*)(Ap + kb * 4);        // global b64, L1-hot
#pragma unroll
            for (int ds_ = 0; ds_ < 4; ++ds_) {
                v2f bf = *(const v2f*)(&Bs[ds_ * 16 + n15][kb * 4 + 2 * half]);
                // (neg_a, A, neg_b, B, c_mod, C, reuse_a, reuse_b)
                acc[ds_] = __builtin_amdgcn_wmma_f32_16x16x4_f32(
                    false, a, false, bf, (short)0, acc[ds_], false, false);
            }
        }

        // distance mask + non-temporal stores
#pragma unroll
        for (int ds_ = 0; ds_ < 4; ++ds_) {
            const int d  = d0 + ds_ * 16 + n15;
            const int b3 = d / GRID_PER_B;
            const int rd = d - b3 * GRID_PER_B;
            const int y3 = 16 + 8 * (rd / G1);
            const int x3 = 16 + 8 * (rd - (rd / G1) * G1);
#pragma unroll
            for (int v = 0; v < 8; ++v) {
                const int m = v + 8 * half;     // C/D layout: VGPR v -> row v (+8 hi half)
                const int qq = qt * 16 + m;
                const int2 xy = xy2ws[qq];
                const int b1 = qq / GRID_PER_B;
                const int dx = x3 - xy.x;
                const int dy = y3 - xy.y;
                const int dis2 = dx * dx + dy * dy + ((b3 != b1) ? 25 : 0); // NEG_D^2
                const float val = (dis2 < 25) ? 0.0f : acc[ds_][v];
                __builtin_nontemporal_store(val, scores + qq * NCOLS + 81 + d);
            }
        }
    }
}

// ---------------------------------------------------------------------------
// Kernel 3: gt matrix — 1.0 in column 0 (npos == 1), 0.0 elsewhere.
// ---------------------------------------------------------------------------
__global__ void __launch_bounds__(256)
gt_fill(float* __restrict__ gt)
{
    const int i = blockIdx.x * 256 + threadIdx.x;
    if (i < NQ * NCOLS) {
        __builtin_nontemporal_store((i % NCOLS) == 0 ? 1.0f : 0.0f, gt + i);
    }
}

extern "C" void kernel_launch(void* const* d_in, const int* in_sizes, int n_in,
                              void* d_out, int out_size, void* d_ws, size_t ws_size,
                              hipStream_t stream)
{
    const float* feat1 = (const float*)d_in[0];
    const float* feat2 = (const float*)d_in[1];
    const float* conf1 = (const float*)d_in[2];
    const float* conf2 = (const float*)d_in[3];
    const float* aflow = (const float*)d_in[4];

    float* scores = (float*)d_out;                       // [NQ][NCOLS]
    float* gt     = scores + (size_t)NQ * NCOLS;         // [NQ][NCOLS]
    float* mask   = gt + (size_t)NQ * NCOLS;             // [NQ]
    float* qconf  = mask + NQ;                           // [NQ]

    float* f1g   = (float*)d_ws;                         // [NQ][128] packed A panel
    float* distg = f1g + (size_t)NQ * CC;                // [NQ][128] packed B panel
    int2*  xy2ws = (int2*)(distg + (size_t)NQ * CC);     // [NQ] target coords

    prep_kernel<<<(NQ + 7) / 8, 256, 0, stream>>>(
        feat1, feat2, conf1, conf2, aflow, f1g, distg, xy2ws, scores, mask, qconf);

    gt_fill<<<(NQ * NCOLS + 255) / 256, 256, 0, stream>>>(gt);

    dscore_wmma<<<QBLK * DBLK, 256, 0, stream>>>(f1g, distg, xy2ws, scores);
}